// ConvTrBlock2d_26783416057883
// MI455X (gfx1250) — compile-verified
//
#include <hip/hip_runtime.h>

typedef float v2f __attribute__((ext_vector_type(2)));
typedef float v8f __attribute__((ext_vector_type(8)));

#define C_IN  64
#define C_OUT 32
#define KK    4
#define WAVES_PER_BLOCK 4
#define ROWS_PER_BLOCK (WAVES_PER_BLOCK * 16)
#define BN_EPS 1e-5f

// d_ws float layout: [0,32) ch-sum  [32,64) ch-sumsq  [64,96) scale  [96,128) shift

__global__ void zero_ws_kernel(float* __restrict__ ws) {
    if (threadIdx.x < 64) ws[threadIdx.x] = 0.0f;
}

// One wave computes a 16-row tile of x against all 4 kernel offsets.
// A fragment (16x4 f32): lane<16 -> row=lane, K = 4s+{0,1}; lane>=16 -> row=lane-16, K = 4s+{2,3}.
// B fragment (4x16 f32): VGPR0: lanes0-15 = W[4s+0][16nt+l], lanes16-31 = W[4s+2][...]; VGPR1: rows 4s+1 / 4s+3.
// C/D (16x16 f32): VGPR j: lanes0-15 -> M=j, lanes16-31 -> M=j+8; N = lane&15.
__global__ __launch_bounds__(128) void gemm_scatter_kernel(
    const float* __restrict__ x, const float* __restrict__ w,
    const int* __restrict__ out_idx, float* __restrict__ y,
    float* __restrict__ ws, int n_in)
{
    __shared__ float fw[KK * C_IN * C_OUT];   // weights, pre-swizzled to B-fragment layout (32 KB)
    __shared__ float lsum[64];                // [0,32) sum, [32,64) sumsq

    const int tid = threadIdx.x;
    if (tid < 64) lsum[tid] = 0.0f;

    // Cooperative swizzled weight load: w[k][cin][cout] ->
    // fw[(((k*2+nt)*16 + s)*32 + lidx)*2 + v], s=cin>>2, v=cin&1, half=(cin>>1)&1,
    // nt=cout>>4, lidx=(cout&15)+16*half. Fragment read is then one b64 per lane.
    for (int f = tid; f < (KK * C_IN * C_OUT) / 4; f += 128) {
        const float4 wv = ((const float4*)w)[f];
        const int e    = f * 4;
        const int k    = e >> 11;         // / (64*32)
        const int rem  = e & 2047;
        const int cin  = rem >> 5;
        const int c0   = rem & 31;
        const int s    = cin >> 2;
        const int v    = cin & 1;
        const int half = (cin >> 1) & 1;
        const float vals[4] = {wv.x, wv.y, wv.z, wv.w};
        #pragma unroll
        for (int d = 0; d < 4; ++d) {
            const int cout = c0 + d;
            const int nt   = cout >> 4;
            const int lidx = (cout & 15) + 16 * half;
            fw[(((k * 2 + nt) * 16 + s) * 32 + lidx) * 2 + v] = vals[d];
        }
    }
    __syncthreads();

    const int lane = tid & 31;
    const int wave = tid >> 5;
    const int lr   = lane & 15;
    const int hf   = lane >> 4;
    const int i0   = (blockIdx.x * WAVES_PER_BLOCK + wave) * 16;

    if (i0 < n_in) {
        // Load the A tile: 16 K-step fragments of 2 floats each.
        v2f a[16];
        const float* xrow = x + (long)(i0 + lr) * C_IN + 2 * hf;
        #pragma unroll
        for (int s = 0; s < 16; ++s)
            a[s] = *(const v2f*)(xrow + 4 * s);

        #pragma unroll
        for (int k = 0; k < KK; ++k) {
            // Scatter targets for this wave's 16 rows at offset k.
            int oidx[8];
            const int* ip = out_idx + (long)k * n_in + i0 + 8 * hf;
            #pragma unroll
            for (int j = 0; j < 8; ++j) oidx[j] = ip[j];

            #pragma unroll
            for (int nt = 0; nt < 2; ++nt) {
                v8f acc = {};
                const float* fb = fw + ((k * 2 + nt) * 16) * 64 + lane * 2;
                #pragma unroll
                for (int s = 0; s < 16; ++s) {
                    v2f b = *(const v2f*)(fb + s * 64);   // ds_load_b64
                    acc = __builtin_amdgcn_wmma_f32_16x16x4_f32(
                        false, a[s], false, b, (short)0, acc, false, false);
                }
                const int col = nt * 16 + lr;
                float psum = 0.0f, psq = 0.0f;
                #pragma unroll
                for (int j = 0; j < 8; ++j) {
                    const float vv = acc[j];
                    y[(long)oidx[j] * C_OUT + col] = vv;   // permutation => plain store
                    psum += vv;
                    psq  += vv * vv;
                }
                atomicAdd(&lsum[col], psum);        // ds_add_f32
                atomicAdd(&lsum[32 + col], psq);
            }
        }
    }
    __syncthreads();
    if (tid < 64) atomicAdd(&ws[tid], lsum[tid]);   // global_atomic_add_f32
}

__global__ void bn_stats_kernel(const float* __restrict__ gamma,
                                const float* __restrict__ beta,
                                float* __restrict__ ws, float inv_n)
{
    const int c = threadIdx.x;
    if (c < 32) {
        const float mean = ws[c] * inv_n;
        const float var  = ws[32 + c] * inv_n - mean * mean;
        const float inv  = rsqrtf(var + BN_EPS);
        const float sc   = gamma[c] * inv;
        ws[64 + c] = sc;
        ws[96 + c] = beta[c] - mean * sc;
    }
}

__global__ __launch_bounds__(256) void bn_relu_kernel(float* __restrict__ y,
                                                      const float* __restrict__ ws,
                                                      long n4)
{
    const long i = (long)blockIdx.x * blockDim.x + threadIdx.x;
    if (i >= n4) return;
    float4 v = ((float4*)y)[i];
    const int c0 = (int)((i * 4) & 31);
    const float4 sc = *(const float4*)(ws + 64 + c0);
    const float4 sh = *(const float4*)(ws + 96 + c0);
    v.x = fmaxf(fmaf(v.x, sc.x, sh.x), 0.0f);
    v.y = fmaxf(fmaf(v.y, sc.y, sh.y), 0.0f);
    v.z = fmaxf(fmaf(v.z, sc.z, sh.z), 0.0f);
    v.w = fmaxf(fmaf(v.w, sc.w, sh.w), 0.0f);
    ((float4*)y)[i] = v;
}

extern "C" void kernel_launch(void* const* d_in, const int* in_sizes, int n_in_args,
                              void* d_out, int out_size, void* d_ws, size_t ws_size,
                              hipStream_t stream) {
    const float* x     = (const float*)d_in[0];   // [N_IN, 64]
    const float* w     = (const float*)d_in[1];   // [4, 64, 32]
    const float* gamma = (const float*)d_in[2];   // [32]
    const float* beta  = (const float*)d_in[3];   // [32]
    const int*   oidx  = (const int*)d_in[4];     // [4, N_IN]
    float*       y     = (float*)d_out;           // [N_OUT, 32]
    float*       ws    = (float*)d_ws;

    const int n_in  = in_sizes[0] / C_IN;
    const int n_out = out_size / C_OUT;
    const float inv_n = 1.0f / (float)n_out;

    zero_ws_kernel<<<1, 64, 0, stream>>>(ws);

    const int gemm_blocks = (n_in + ROWS_PER_BLOCK - 1) / ROWS_PER_BLOCK;
    gemm_scatter_kernel<<<gemm_blocks, 128, 0, stream>>>(x, w, oidx, y, ws, n_in);

    bn_stats_kernel<<<1, 32, 0, stream>>>(gamma, beta, ws, inv_n);

    const long n4 = (long)out_size / 4;
    const int  bn_blocks = (int)((n4 + 255) / 256);
    bn_relu_kernel<<<bn_blocks, 256, 0, stream>>>(y, ws, n4);
}